// BalancedCELoss_46729244180707
// MI455X (gfx1250) — compile-verified
//
#include <hip/hip_runtime.h>
#include <hip/hip_bf16.h>

// ---------------------------------------------------------------------------
// BalancedCELoss on MI455X (gfx1250).
//  - Main pass: 1 streaming read of probs [2,16,96,96,96] f32.
//  - Per-voxel class dot products done with chained V_WMMA_F32_16X16X4_F32:
//      D = W(16x16 weights) x P(16 classes x 16 voxels), K=16 via 4 WMMAs.
//    acc1 rows: 0=wz(z-slice),1=wy(y-slice),2=w_dense ; acc2 rows: Wx, diag.
//  - Entropy term computed on the same B-operand registers (each prob read once).
// ---------------------------------------------------------------------------

typedef float v2f __attribute__((ext_vector_type(2)));
typedef float v8f __attribute__((ext_vector_type(8)));

#define EPSF 1e-6f
#define ZYX 884736
#define YX  9216

__device__ __forceinline__ float clampp(float x) {
    return fminf(fmaxf(x, EPSF), 1.0f - EPSF);
}
__device__ __forceinline__ float focalf(float x) {
    float om = 1.0f - x;                       // gamma = 2
    return -om * om * __logf(clampp(x));
}

// ---------------- workspace layout (word offsets from d_ws) ----------------
// float side:
//   [0..15]    accum[b][j]  j: 0=sZ 1=sY 2=sX 3=dense 4=regSum
//   [16..207]  actz[2][96]  [208..399] acty  [400..591] actx
//   [592..623] wdense[2][16]
// int side (word offset 1024):
//   [0..191] hz  [192..383] hy  [384..575] hx
//   [576..767] vz  [768..959] msum_y  [960..1151] msum_x  [1152..1153] anynz

__global__ __launch_bounds__(256)
void bce_init(int* ws) {
    for (int i = threadIdx.x; i < 2304; i += 256) ws[i] = 0;
}

// One block per (b,z) plane: foreground/valid stats from target & masks.
__global__ __launch_bounds__(256)
void bce_stats(const int* __restrict__ target, const int* __restrict__ masks,
               int* __restrict__ wsI) {
    int b = blockIdx.x / 96, z = blockIdx.x % 96;
    __shared__ int fgy[96], fgx[96], my[96], mx[96];
    __shared__ int fgany, msum;
    if (threadIdx.x < 96) { fgy[threadIdx.x] = 0; fgx[threadIdx.x] = 0;
                            my[threadIdx.x] = 0;  mx[threadIdx.x] = 0; }
    if (threadIdx.x == 0) { fgany = 0; msum = 0; }
    __syncthreads();
    int anyl = 0, msl = 0;
    size_t base = (size_t)b * ZYX + (size_t)z * YX;
    for (int i = threadIdx.x; i < YX; i += 256) {
        int y = i / 96, x = i - y * 96;
        int t = target[base + i];
        int m = masks[base + i];
        if (t > 0) { anyl = 1; atomicOr(&fgy[y], 1); atomicOr(&fgx[x], 1); }
        msl += m; atomicAdd(&my[y], m); atomicAdd(&mx[x], m);
    }
    if (anyl) atomicOr(&fgany, 1);
    atomicAdd(&msum, msl);
    __syncthreads();
    if (threadIdx.x < 96) {
        atomicOr (&wsI[192 + b * 96 + threadIdx.x], fgy[threadIdx.x]);
        atomicOr (&wsI[384 + b * 96 + threadIdx.x], fgx[threadIdx.x]);
        atomicAdd(&wsI[768 + b * 96 + threadIdx.x], my[threadIdx.x]);
        atomicAdd(&wsI[960 + b * 96 + threadIdx.x], mx[threadIdx.x]);
    }
    if (threadIdx.x == 0) {
        wsI[b * 96 + z]       = fgany;                 // hz: one owner block
        wsI[576 + b * 96 + z] = (msum == YX) ? 1 : 0;  // vz
        if (fgany) atomicOr(&wsI[1152 + b], 1);        // anynz
    }
}

// act arrays + dense class weights.
__global__ __launch_bounds__(256)
void bce_prep(const int* __restrict__ afc, const int* __restrict__ wsI,
              float* __restrict__ wsF) {
    int i = threadIdx.x;
    if (i < 192) {
        wsF[16  + i] = (wsI[i]       && wsI[576 + i])        ? 1.f : 0.f; // actz
        wsF[208 + i] = (wsI[192 + i] && (wsI[768 + i] == YX)) ? 1.f : 0.f; // acty
        wsF[400 + i] = (wsI[384 + i] && (wsI[960 + i] == YX)) ? 1.f : 0.f; // actx
    }
    if (i < 32) {
        int b = i >> 4, c = i & 15, annot = 0;
        for (int j = 0; j < 16; j++) {
            int v = afc[b * 16 + j];
            if (v == c && v > 0) annot = 1;
        }
        wsF[592 + i] = annot ? 0.f : 1.f;  // dense weight: unannotated classes
    }
}

// Main streaming pass. One wave per (b,z,y) row; 6 tiles of 16 voxels in x.
__global__ __launch_bounds__(256)
void bce_main(const float* __restrict__ probs, const int* __restrict__ target,
              const int* __restrict__ az, const int* __restrict__ ay,
              const int* __restrict__ ax, const float* __restrict__ wsF,
              float* __restrict__ accum) {
    const float* actz = wsF + 16;
    const float* acty = wsF + 208;
    const float* actx = wsF + 400;
    const float* wdw  = wsF + 592;

    int wave = threadIdx.x >> 5, lane = threadIdx.x & 31;
    int row  = blockIdx.x * 8 + wave;             // b*Z*Y + z*Y + y
    int b = row / 9216; int rem = row - b * 9216;
    int z = rem / 96;   int y   = rem - z * 96;
    int ln = lane & 15, lh = lane >> 4;

    const float* pbase = probs + (size_t)b * 16 * ZYX + (size_t)z * YX + y * 96;
    const int*   tbase = target + (size_t)b * ZYX + (size_t)z * YX + y * 96;

    // Chain-1 A matrix (weights, 16x4 per step): row0=wz, row1=wy, row2=wdense.
    // A layout: lane m holds A[m][2*lh+{0,1}] for class block 4k.
    v2f a1[4];
#pragma unroll
    for (int k = 0; k < 4; k++) {
        int c0 = 4 * k + 2 * lh;
        float w0 = 0.f, w1 = 0.f;
        if (ln == 0) {
            w0 = (az[(b * 96 + z) * 16 + c0]     <= 0) ? 1.f : 0.f;
            w1 = (az[(b * 96 + z) * 16 + c0 + 1] <= 0) ? 1.f : 0.f;
        } else if (ln == 1) {
            w0 = (ay[(b * 96 + y) * 16 + c0]     <= 0) ? 1.f : 0.f;
            w1 = (ay[(b * 96 + y) * 16 + c0 + 1] <= 0) ? 1.f : 0.f;
        } else if (ln == 2) {
            w0 = wdw[b * 16 + c0]; w1 = wdw[b * 16 + c0 + 1];
        }
        a1[k] = (v2f){w0, w1};
    }
    float actzv = actz[b * 96 + z];
    float actyv = acty[b * 96 + y];
    float vmask = (lane < 16) ? 1.f : 0.f;
    float aZ = 0.f, aY = 0.f, aX = 0.f, aD = 0.f, aE = 0.f;

    for (int tx = 0; tx < 6; tx++) {
        int x0 = tx * 16;
        // B operand: probs tile, B_k[kk][n] = p[4k+kk][x0+n]; lane n holds
        // rows {2*lh, 2*lh+1}. Coalesced 64B runs per class row.
        v2f bk[4];
#pragma unroll
        for (int k = 0; k < 4; k++) {
            int c0 = 4 * k + 2 * lh;
            bk[k].x = pbase[(size_t)c0 * ZYX + x0 + ln];
            bk[k].y = pbase[(size_t)(c0 + 1) * ZYX + x0 + ln];
        }
        // Chain-2 A matrix: row m = wx[x0+m][:] (diagonal of D2 = per-voxel x-dot)
        v2f a2[4];
#pragma unroll
        for (int k = 0; k < 4; k++) {
            int c0   = 4 * k + 2 * lh;
            int base = (b * 96 + x0 + ln) * 16;
            a2[k].x = (ax[base + c0]     <= 0) ? 1.f : 0.f;
            a2[k].y = (ax[base + c0 + 1] <= 0) ? 1.f : 0.f;
        }
        v8f acc1 = {}; v8f acc2 = {};
#pragma unroll
        for (int k = 0; k < 4; k++) {
            acc1 = __builtin_amdgcn_wmma_f32_16x16x4_f32(
                false, a1[k], false, bk[k], (short)0, acc1, false, false);
            acc2 = __builtin_amdgcn_wmma_f32_16x16x4_f32(
                false, a2[k], false, bk[k], (short)0, acc2, false, false);
        }
        // Entropy partial: every prob of the tile appears exactly once per wave.
#pragma unroll
        for (int k = 0; k < 4; k++) {
            aE += bk[k].x * __logf(clampp(bk[k].x));
            aE += bk[k].y * __logf(clampp(bk[k].y));
        }
        // x-dot diagonal: D2[v][v] lives at (vgpr v&7, lane v<8 ? v : v+16).
        int   i8   = lane & 7;
        float cand = (i8 & 4) ? ((i8 & 2) ? ((i8 & 1) ? acc2[7] : acc2[6])
                                          : ((i8 & 1) ? acc2[5] : acc2[4]))
                              : ((i8 & 2) ? ((i8 & 1) ? acc2[3] : acc2[2])
                                          : ((i8 & 1) ? acc2[1] : acc2[0]));
        cand = ((lane < 8) || (lane >= 24)) ? cand : 0.f;
        int   src = (ln < 8) ? ln : ln + 16;
        float sx  = __shfl(cand, src, 32);
        // Columns of D1 land in the voxel's own lane (lanes 0..15).
        float sz0 = acc1[0], sy0 = acc1[1], sd0 = acc1[2];

        int   t    = tbase[x0 + ln];
        float pt   = pbase[(size_t)t * ZYX + x0 + ln];
        bool  fg   = t > 0;
        float cefg = focalf(pt);
        float cez = fg ? cefg : focalf(sz0);
        float cey = fg ? cefg : focalf(sy0);
        float cex = fg ? cefg : focalf(sx);
        float ced = fg ? cefg : focalf(sd0);
        float axv = actx[b * 96 + x0 + ln];
        aZ += cez * actzv * vmask;
        aY += cey * actyv * vmask;
        aX += cex * axv   * vmask;
        aD += ced * vmask;
    }

    // Block reduction (same b for all 8 waves of a block) + 5 atomics.
    __shared__ float red[256];
    float vals[5] = {aZ, aY, aX, aD, aE};
    for (int j = 0; j < 5; j++) {
        red[threadIdx.x] = vals[j];
        __syncthreads();
        for (int s = 128; s > 0; s >>= 1) {
            if (threadIdx.x < s) red[threadIdx.x] += red[threadIdx.x + s];
            __syncthreads();
        }
        if (threadIdx.x == 0) atomicAdd(&accum[b * 8 + j], red[0]);
        __syncthreads();
    }
}

__global__ void bce_final(const float* __restrict__ wsF, const int* __restrict__ wsI,
                          const int* __restrict__ is_sparse, float* __restrict__ out) {
    if (threadIdx.x != 0 || blockIdx.x != 0) return;
    const float* accum = wsF;
    const float* actz = wsF + 16;
    const float* acty = wsF + 208;
    const float* actx = wsF + 400;
    float ce[2], has[2], reg[2];
    for (int b = 0; b < 2; b++) {
        float cz = 0.f, cy = 0.f, cx = 0.f;
        for (int i = 0; i < 96; i++) {
            cz += actz[b * 96 + i]; cy += acty[b * 96 + i]; cx += actx[b * 96 + i];
        }
        float cntz = cz * 9216.f, cnty = cy * 9216.f, cntx = cx * 9216.f;
        float mz = accum[b * 8 + 0] / fmaxf(cntz, 1.f);
        float my = accum[b * 8 + 1] / fmaxf(cnty, 1.f);
        float mx = accum[b * 8 + 2] / fmaxf(cntx, 1.f);
        float bz = cntz > 0.f ? 1.f : 0.f;
        float by = cnty > 0.f ? 1.f : 0.f;
        float bx = cntx > 0.f ? 1.f : 0.f;
        float nax    = bz + by + bx;
        float sparse = (mz * bz + my * by + mx * bx) / fmaxf(nax, 1.f);
        float dense  = accum[b * 8 + 3] / 884736.f;
        float ent    = -accum[b * 8 + 4] / 884736.f;
        int   sp     = is_sparse[b];           // [B,1] flat
        ce[b]  = (sp == 1) ? sparse : dense;
        has[b] = (sp == 1) ? ((nax > 0.f) ? 1.f : 0.f) : 1.f;
        reg[b] = wsI[1152 + b] ? ent : 3.f * ent;   // MULT=3 if all targets zero
    }
    float n = has[0] + has[1];
    out[0] = (n > 0.f) ? (ce[0] * has[0] + ce[1] * has[1]) / fmaxf(n, 1.f) : 0.f;
    out[1] = 0.5f * (reg[0] + reg[1]);
}

extern "C" void kernel_launch(void* const* d_in, const int* in_sizes, int n_in,
                              void* d_out, int out_size, void* d_ws, size_t ws_size,
                              hipStream_t stream) {
    const float* probs  = (const float*)d_in[0];
    const int*   target = (const int*)d_in[1];
    const int*   afc    = (const int*)d_in[2];
    const int*   az     = (const int*)d_in[3];
    const int*   ay     = (const int*)d_in[4];
    const int*   ax     = (const int*)d_in[5];
    const int*   masks  = (const int*)d_in[6];
    const int*   is_sp  = (const int*)d_in[7];
    float* out = (float*)d_out;

    float* wsF = (float*)d_ws;
    int*   wsI = (int*)d_ws + 1024;

    bce_init <<<1, 256, 0, stream>>>((int*)d_ws);
    bce_stats<<<192, 256, 0, stream>>>(target, masks, wsI);
    bce_prep <<<1, 256, 0, stream>>>(afc, wsI, wsF);
    bce_main <<<2304, 256, 0, stream>>>(probs, target, az, ay, ax, wsF, wsF /*accum*/);
    bce_final<<<1, 32, 0, stream>>>(wsF, wsI, is_sp, out);
}